// SpatialAttn2_59210419142959
// MI455X (gfx1250) — compile-verified
//
#include <hip/hip_runtime.h>
#include <hip/hip_bf16.h>

typedef float v2f __attribute__((ext_vector_type(2)));
typedef float v8f __attribute__((ext_vector_type(8)));

#define BB 16
#define LL 32
#define NN 256
#define PP 16
#define HH 4
#define FF 12
#define WWIN 16
#define HIDD 4

// LDS partition (floats), dynamic shared:
//  Xs  [256*17]            @ 0
//  Ys  [256*17]            @ 4352
//  Was [16*16]             @ 8704
//  slab[8][16*260]         @ 8960   (per-wave, rows 16B-aligned)
//  Zs  [8][16*17]          @ 42240  (per-wave)
#define XS_OFF   0
#define YS_OFF   4352
#define WAS_OFF  8704
#define SLAB_OFF 8960
#define SLAB_ROW 260    // multiple of 4 -> float4 rows; 260*l % 64 = 4*l (conflict-free)
#define SLAB_STR (16 * SLAB_ROW)
#define ZS_OFF   42240
#define ZS_STR   272    // 16*17
#define SMEM_FLOATS 44416

__device__ __forceinline__ v8f wmma_f32_k4(v2f a, v2f b, v8f c) {
    return __builtin_amdgcn_wmma_f32_16x16x4_f32(false, a, false, b, (short)0, c, false, false);
}

__device__ __forceinline__ float fmax8(const float4& a, const float4& b) {
    return fmaxf(fmaxf(fmaxf(a.x, a.y), fmaxf(a.z, a.w)),
                 fmaxf(fmaxf(b.x, b.y), fmaxf(b.z, b.w)));
}

// ---------------------------------------------------------------------------
// Layer 1: one block per (b, w, h).  X: [N,P] from sliding window of inp.
// score1 = (X @ Wa1[h]) @ X^T * scale ; masked softmax -> attn1 (global)
// Z = attn1 @ X ; H1[:, h*4:(h+1)*4] = Z @ Wv1[h]  -> workspace
// ---------------------------------------------------------------------------
__global__ void attn_layer1(const float* __restrict__ inp,
                            const int*   __restrict__ mask,
                            const float* __restrict__ Wa1,
                            const float* __restrict__ Wv1,
                            float* __restrict__ attn1_out,
                            float* __restrict__ h1_out) {
    extern __shared__ float sm[];
    const int h    = blockIdx.x & (HH - 1);
    const int bw   = blockIdx.x >> 2;          // b*W + w
    const int b    = bw >> 4;
    const int w    = bw & (WWIN - 1);
    const int tid  = threadIdx.x;
    const int lane = tid & 31;
    const int wv   = tid >> 5;
    const int half = lane >> 4;                // 0 | 1
    const int l16  = lane & 15;

    float* Xs   = sm + XS_OFF;
    float* Ys   = sm + YS_OFF;
    float* Was  = sm + WAS_OFF;
    float* slab = sm + SLAB_OFF + wv * SLAB_STR;
    float* Zs   = sm + ZS_OFF + wv * ZS_STR;

    // X[n][p] = inp[b, w+p, n]  -- b128 loads along n, scatter to stride-17 LDS
    {
        const int p  = tid & 15;
        const int g0 = (tid >> 4) * 4;
        const float4* src = (const float4*)(inp + ((size_t)b * LL + w + p) * NN);
        #pragma unroll
        for (int gg = 0; gg < 4; ++gg) {
            const int g = g0 + gg;
            const float4 v = src[g];
            Xs[(4 * g + 0) * 17 + p] = v.x;
            Xs[(4 * g + 1) * 17 + p] = v.y;
            Xs[(4 * g + 2) * 17 + p] = v.z;
            Xs[(4 * g + 3) * 17 + p] = v.w;
        }
    }
    Was[tid & 255] = Wa1[h * 256 + (tid & 255)];
    __syncthreads();

    // ---- Y = X @ Wa1[h] : 16 row tiles, 2 per wave
    #pragma unroll
    for (int j = 0; j < 2; ++j) {
        const int rowbase = (wv * 2 + j) * 16;
        v8f acc = {};
        #pragma unroll
        for (int k0 = 0; k0 < 16; k0 += 4) {
            const int k = k0 + 2 * half;
            v2f a, bf;
            a.x  = Xs[(rowbase + l16) * 17 + k];
            a.y  = Xs[(rowbase + l16) * 17 + k + 1];
            bf.x = Was[k * 16 + l16];
            bf.y = Was[(k + 1) * 16 + l16];
            acc = wmma_f32_k4(a, bf, acc);
        }
        #pragma unroll
        for (int r = 0; r < 8; ++r)
            Ys[(rowbase + r + 8 * half) * 17 + l16] = acc[r];
    }
    __syncthreads();

    const float scale = 0.25f;  // 1/sqrt(P)
    const size_t a1base = ((size_t)bw * HH + h) * NN * NN;

    for (int j = 0; j < 2; ++j) {
        const int rowbase = (wv * 2 + j) * 16;

        // ---- score slab: 16 rows x 256 cols, 16 col-tiles of chained WMMA
        for (int t = 0; t < 16; ++t) {
            const int colbase = t * 16;
            v8f acc = {};
            #pragma unroll
            for (int k0 = 0; k0 < 16; k0 += 4) {
                const int k = k0 + 2 * half;
                v2f a, bf;
                a.x  = Ys[(rowbase + l16) * 17 + k];
                a.y  = Ys[(rowbase + l16) * 17 + k + 1];
                bf.x = Xs[(colbase + l16) * 17 + k];   // B[k][n] = X[n][k]
                bf.y = Xs[(colbase + l16) * 17 + k + 1];
                acc = wmma_f32_k4(a, bf, acc);
            }
            #pragma unroll
            for (int r = 0; r < 8; ++r)
                slab[(r + 8 * half) * SLAB_ROW + colbase + l16] = acc[r] * scale;
        }

        // ---- masked softmax per row: float4 lanes, wave32 shfl reductions
        for (int r = 0; r < 16; ++r) {
            const int nrow = rowbase + r;
            float4 va = ((const float4*)(slab + r * SLAB_ROW))[lane];
            float4 vb = ((const float4*)(slab + r * SLAB_ROW + 128))[lane];
            const int4 ma = ((const int4*)(mask + (size_t)nrow * NN))[lane];
            const int4 mb = ((const int4*)(mask + (size_t)nrow * NN + 128))[lane];
            va.x = (ma.x > 0) ? va.x : -1e9f;
            va.y = (ma.y > 0) ? va.y : -1e9f;
            va.z = (ma.z > 0) ? va.z : -1e9f;
            va.w = (ma.w > 0) ? va.w : -1e9f;
            vb.x = (mb.x > 0) ? vb.x : -1e9f;
            vb.y = (mb.y > 0) ? vb.y : -1e9f;
            vb.z = (mb.z > 0) ? vb.z : -1e9f;
            vb.w = (mb.w > 0) ? vb.w : -1e9f;
            float mx = fmax8(va, vb);
            #pragma unroll
            for (int off = 16; off > 0; off >>= 1)
                mx = fmaxf(mx, __shfl_xor(mx, off, 32));
            va.x = __expf(va.x - mx); va.y = __expf(va.y - mx);
            va.z = __expf(va.z - mx); va.w = __expf(va.w - mx);
            vb.x = __expf(vb.x - mx); vb.y = __expf(vb.y - mx);
            vb.z = __expf(vb.z - mx); vb.w = __expf(vb.w - mx);
            float sum = (va.x + va.y) + (va.z + va.w) + (vb.x + vb.y) + (vb.z + vb.w);
            #pragma unroll
            for (int off = 16; off > 0; off >>= 1)
                sum += __shfl_xor(sum, off, 32);
            const float inv = 1.f / sum;
            va.x *= inv; va.y *= inv; va.z *= inv; va.w *= inv;
            vb.x *= inv; vb.y *= inv; vb.z *= inv; vb.w *= inv;
            ((float4*)(slab + r * SLAB_ROW))[lane]       = va;
            ((float4*)(slab + r * SLAB_ROW + 128))[lane] = vb;
            ((float4*)(attn1_out + a1base + (size_t)nrow * NN))[lane]       = va;
            ((float4*)(attn1_out + a1base + (size_t)nrow * NN + 128))[lane] = vb;
        }

        // ---- Z = P @ X : K = 256 chained WMMA
        {
            v8f acc = {};
            for (int k0 = 0; k0 < NN; k0 += 4) {
                const int k = k0 + 2 * half;
                v2f a, bf;
                a.x  = slab[l16 * SLAB_ROW + k];
                a.y  = slab[l16 * SLAB_ROW + k + 1];
                bf.x = Xs[k * 17 + l16];
                bf.y = Xs[(k + 1) * 17 + l16];
                acc = wmma_f32_k4(a, bf, acc);
            }
            #pragma unroll
            for (int r = 0; r < 8; ++r)
                Zs[(r + 8 * half) * 17 + l16] = acc[r];
        }

        // ---- H1 head slice = Z @ Wv1[h]  (16x16 @ 16x4, tiny -> VALU)
        for (int idx = lane; idx < 64; idx += 32) {   // exactly 2 uniform iters
            const int m = idx >> 2, o = idx & 3;
            float s = 0.f;
            #pragma unroll
            for (int p = 0; p < PP; ++p)
                s += Zs[m * 17 + p] * Wv1[h * (PP * HIDD) + p * HIDD + o];
            h1_out[((size_t)bw * NN + rowbase + m) * PP + h * HIDD + o] = s;
        }
    }
}

// ---------------------------------------------------------------------------
// Layer 2: one block per (b, w).  Same pipeline on H1 with Wa2 / Wv2.
// Writes attn2 and out[b,w,f,n] = h2[n][f].
// ---------------------------------------------------------------------------
__global__ void attn_layer2(const float* __restrict__ h1,
                            const int*   __restrict__ mask,
                            const float* __restrict__ Wa2,
                            const float* __restrict__ Wv2,
                            float* __restrict__ attn2_out,
                            float* __restrict__ out) {
    extern __shared__ float sm[];
    const int bw   = blockIdx.x;
    const int tid  = threadIdx.x;
    const int lane = tid & 31;
    const int wv   = tid >> 5;
    const int half = lane >> 4;
    const int l16  = lane & 15;

    float* Xs   = sm + XS_OFF;     // holds H1 here
    float* Ys   = sm + YS_OFF;
    float* Was  = sm + WAS_OFF;
    float* slab = sm + SLAB_OFF + wv * SLAB_STR;
    float* Zs   = sm + ZS_OFF + wv * ZS_STR;

    {
        const int n = tid;
        const float4* src = (const float4*)(h1 + ((size_t)bw * NN + n) * PP);
        #pragma unroll
        for (int q = 0; q < 4; ++q) {
            const float4 v = src[q];
            Xs[n * 17 + 4 * q + 0] = v.x;
            Xs[n * 17 + 4 * q + 1] = v.y;
            Xs[n * 17 + 4 * q + 2] = v.z;
            Xs[n * 17 + 4 * q + 3] = v.w;
        }
    }
    Was[tid & 255] = Wa2[tid & 255];
    __syncthreads();

    #pragma unroll
    for (int j = 0; j < 2; ++j) {
        const int rowbase = (wv * 2 + j) * 16;
        v8f acc = {};
        #pragma unroll
        for (int k0 = 0; k0 < 16; k0 += 4) {
            const int k = k0 + 2 * half;
            v2f a, bf;
            a.x  = Xs[(rowbase + l16) * 17 + k];
            a.y  = Xs[(rowbase + l16) * 17 + k + 1];
            bf.x = Was[k * 16 + l16];
            bf.y = Was[(k + 1) * 16 + l16];
            acc = wmma_f32_k4(a, bf, acc);
        }
        #pragma unroll
        for (int r = 0; r < 8; ++r)
            Ys[(rowbase + r + 8 * half) * 17 + l16] = acc[r];
    }
    __syncthreads();

    const float scale = 0.25f;
    const size_t a2base = (size_t)bw * NN * NN;

    for (int j = 0; j < 2; ++j) {
        const int rowbase = (wv * 2 + j) * 16;

        for (int t = 0; t < 16; ++t) {
            const int colbase = t * 16;
            v8f acc = {};
            #pragma unroll
            for (int k0 = 0; k0 < 16; k0 += 4) {
                const int k = k0 + 2 * half;
                v2f a, bf;
                a.x  = Ys[(rowbase + l16) * 17 + k];
                a.y  = Ys[(rowbase + l16) * 17 + k + 1];
                bf.x = Xs[(colbase + l16) * 17 + k];
                bf.y = Xs[(colbase + l16) * 17 + k + 1];
                acc = wmma_f32_k4(a, bf, acc);
            }
            #pragma unroll
            for (int r = 0; r < 8; ++r)
                slab[(r + 8 * half) * SLAB_ROW + colbase + l16] = acc[r] * scale;
        }

        for (int r = 0; r < 16; ++r) {
            const int nrow = rowbase + r;
            float4 va = ((const float4*)(slab + r * SLAB_ROW))[lane];
            float4 vb = ((const float4*)(slab + r * SLAB_ROW + 128))[lane];
            const int4 ma = ((const int4*)(mask + (size_t)nrow * NN))[lane];
            const int4 mb = ((const int4*)(mask + (size_t)nrow * NN + 128))[lane];
            va.x = (ma.x > 0) ? va.x : -1e9f;
            va.y = (ma.y > 0) ? va.y : -1e9f;
            va.z = (ma.z > 0) ? va.z : -1e9f;
            va.w = (ma.w > 0) ? va.w : -1e9f;
            vb.x = (mb.x > 0) ? vb.x : -1e9f;
            vb.y = (mb.y > 0) ? vb.y : -1e9f;
            vb.z = (mb.z > 0) ? vb.z : -1e9f;
            vb.w = (mb.w > 0) ? vb.w : -1e9f;
            float mx = fmax8(va, vb);
            #pragma unroll
            for (int off = 16; off > 0; off >>= 1)
                mx = fmaxf(mx, __shfl_xor(mx, off, 32));
            va.x = __expf(va.x - mx); va.y = __expf(va.y - mx);
            va.z = __expf(va.z - mx); va.w = __expf(va.w - mx);
            vb.x = __expf(vb.x - mx); vb.y = __expf(vb.y - mx);
            vb.z = __expf(vb.z - mx); vb.w = __expf(vb.w - mx);
            float sum = (va.x + va.y) + (va.z + va.w) + (vb.x + vb.y) + (vb.z + vb.w);
            #pragma unroll
            for (int off = 16; off > 0; off >>= 1)
                sum += __shfl_xor(sum, off, 32);
            const float inv = 1.f / sum;
            va.x *= inv; va.y *= inv; va.z *= inv; va.w *= inv;
            vb.x *= inv; vb.y *= inv; vb.z *= inv; vb.w *= inv;
            ((float4*)(slab + r * SLAB_ROW))[lane]       = va;
            ((float4*)(slab + r * SLAB_ROW + 128))[lane] = vb;
            ((float4*)(attn2_out + a2base + (size_t)nrow * NN))[lane]       = va;
            ((float4*)(attn2_out + a2base + (size_t)nrow * NN + 128))[lane] = vb;
        }

        {
            v8f acc = {};
            for (int k0 = 0; k0 < NN; k0 += 4) {
                const int k = k0 + 2 * half;
                v2f a, bf;
                a.x  = slab[l16 * SLAB_ROW + k];
                a.y  = slab[l16 * SLAB_ROW + k + 1];
                bf.x = Xs[k * 17 + l16];
                bf.y = Xs[(k + 1) * 17 + l16];
                acc = wmma_f32_k4(a, bf, acc);
            }
            #pragma unroll
            for (int r = 0; r < 8; ++r)
                Zs[(r + 8 * half) * 17 + l16] = acc[r];
        }

        // h2 = Z @ Wv2 (16x16 @ 16x12), store transposed: out[bw, f, n]
        for (int idx = lane; idx < 192; idx += 32) {  // exactly 6 uniform iters
            const int m = idx / FF, f = idx % FF;
            float s = 0.f;
            #pragma unroll
            for (int p = 0; p < PP; ++p)
                s += Zs[m * 17 + p] * Wv2[p * FF + f];
            out[((size_t)bw * FF + f) * NN + rowbase + m] = s;
        }
    }
}

__global__ void copy_tail(const float* __restrict__ Wv1, const float* __restrict__ Wv2,
                          float* __restrict__ d1, float* __restrict__ d2) {
    const int t = threadIdx.x;
    if (t < HH * PP * HIDD) d1[t] = Wv1[t];
    if (t < PP * FF)        d2[t] = Wv2[t];
}

extern "C" void kernel_launch(void* const* d_in, const int* in_sizes, int n_in,
                              void* d_out, int out_size, void* d_ws, size_t ws_size,
                              hipStream_t stream) {
    (void)in_sizes; (void)n_in; (void)out_size; (void)ws_size;
    const float* inp  = (const float*)d_in[0];
    const int*   mask = (const int*)d_in[1];
    const float* Wa1  = (const float*)d_in[2];
    const float* Wv1  = (const float*)d_in[3];
    const float* Wa2  = (const float*)d_in[4];
    const float* Wv2  = (const float*)d_in[5];

    float* out = (float*)d_out;
    const size_t n_out   = (size_t)BB * WWIN * FF * NN;          //   786,432
    const size_t n_attn1 = (size_t)BB * WWIN * HH * NN * NN;     // 67,108,864
    const size_t n_attn2 = (size_t)BB * WWIN * NN * NN;          // 16,777,216
    float* out_main = out;
    float* attn1    = out + n_out;
    float* attn2    = out + n_out + n_attn1;
    float* wv1o     = out + n_out + n_attn1 + n_attn2;
    float* wv2o     = wv1o + (size_t)HH * PP * HIDD;

    float* h1ws = (float*)d_ws;  // [B*W, N, P] = 4 MB

    const size_t smem = SMEM_FLOATS * sizeof(float);  // ~174 KB (<320 KB/WGP)
    hipFuncSetAttribute((const void*)attn_layer1,
                        hipFuncAttributeMaxDynamicSharedMemorySize, (int)smem);
    hipFuncSetAttribute((const void*)attn_layer2,
                        hipFuncAttributeMaxDynamicSharedMemorySize, (int)smem);

    attn_layer1<<<dim3(BB * WWIN * HH), dim3(256), smem, stream>>>(
        inp, mask, Wa1, Wv1, attn1, h1ws);
    attn_layer2<<<dim3(BB * WWIN), dim3(256), smem, stream>>>(
        h1ws, mask, Wa2, Wv2, attn2, out_main);
    copy_tail<<<dim3(1), dim3(256), 0, stream>>>(Wv1, Wv2, wv1o, wv2o);
}